// NT_XentLoss_5171140624722
// MI455X (gfx1250) — compile-verified
//
#include <hip/hip_runtime.h>
#include <hip/hip_bf16.h>

typedef __attribute__((ext_vector_type(2))) float v2f;
typedef __attribute__((ext_vector_type(8))) float v8f;

#define NROWS 8192      // 2*B
#define DDIM  128
#define NTILES 512      // NROWS/16
#define CHUNKS 8
#define TILES_PER_CHUNK 64   // NTILES/CHUNKS

// ---------------------------------------------------------------------------
// Kernel 1: L2-normalize rows of zis/zjs into rep[8192][128]; zero rowsum.
// One wave (32 lanes) per row; each lane owns 4 consecutive floats.
// ---------------------------------------------------------------------------
__global__ void ntx_prep(const float* __restrict__ zis,
                         const float* __restrict__ zjs,
                         float* __restrict__ rep,
                         float* __restrict__ rowsum) {
    const int tid  = blockIdx.x * blockDim.x + threadIdx.x;
    const int lane = threadIdx.x & 31;
    const int row  = tid >> 5;

    if (row < NROWS) {
        const float* src = (row < NROWS / 2)
                         ? zis + (size_t)row * DDIM
                         : zjs + (size_t)(row - NROWS / 2) * DDIM;
        float4 x = *(const float4*)(src + lane * 4);
        float ss = x.x * x.x + x.y * x.y + x.z * x.z + x.w * x.w;
        #pragma unroll
        for (int off = 16; off >= 1; off >>= 1)
            ss += __shfl_xor(ss, off, 32);
        const float denom = fmaxf(sqrtf(ss), 1e-12f);
        const float inv = 1.0f / denom;
        float4 y = {x.x * inv, x.y * inv, x.z * inv, x.w * inv};
        *(float4*)(rep + (size_t)row * DDIM + lane * 4) = y;
    }
    if (tid < NROWS) rowsum[tid] = 0.0f;
}

// ---------------------------------------------------------------------------
// Kernel 2: fused sim = (rep·repT)*2 with streaming exp-sum per row.
// grid = (64 row-blocks, 8 column-chunks), block = 256 threads = 8 waves.
// Wave w owns rows r0 = (blockIdx.x*8 + w)*16 (one WMMA M-tile).
// A operands live in VGPRs (ISA 16x4 f32 A layout). B operands are staged
// into a double-buffered LDS tile in exact WMMA lane order: next tile's
// global loads are issued into registers before the WMMA block (latency
// hidden under matrix math), then stored to the alternate buffer.
// Two independent WMMA accumulation chains keep the XDL pipe fed.
// ---------------------------------------------------------------------------
__global__ __launch_bounds__(256) void ntx_main(const float* __restrict__ rep,
                                                float* __restrict__ rowsum,
                                                float* __restrict__ posArr) {
    __shared__ v2f lb[2][32 * 32];   // double-buffered B operands (2 x 8 KB)

    const int tx   = (int)threadIdx.x;
    const int lane = tx & 31;
    const int wave = tx >> 5;
    const int rt   = blockIdx.x * 8 + wave;     // row-tile index 0..511
    const int r0   = rt * 16;
    const int chunk = blockIdx.y;

    const int m  = lane & 15;                   // A row within tile
    const int kh = (lane & 16) ? 2 : 0;         // K sub-offset per ISA layout

    // Preload A: 32 k-chunks of float2 for this wave's 16 rows.
    v2f a[32];
    const float* Arow = rep + (size_t)(r0 + m) * DDIM;
    #pragma unroll
    for (int j = 0; j < 32; ++j)
        a[j] = *(const v2f*)(Arow + 4 * j + kh);

    float acc[8];
    #pragma unroll
    for (int v = 0; v < 8; ++v) acc[v] = 0.0f;

    const int posTile = (rt + 256) & (NTILES - 1);
    const int tbase   = chunk * TILES_PER_CHUNK;

    // Per-thread staging slots: lb[buf][e] = rep[c0 + e%16][4*(e/32) + ((e&16)?2:0)]
    const int e_j[4]  = { tx >> 5, (tx + 256) >> 5, (tx + 512) >> 5, (tx + 768) >> 5 };
    const int e_col   = tx & 15;
    const int e_kk    = (tx & 16) ? 2 : 0;

    v2f stage[4];

    // Prologue: fill buffer 0 with the first tile.
    {
        const size_t cb = (size_t)(tbase * 16 + e_col) * DDIM + e_kk;
        #pragma unroll
        for (int i = 0; i < 4; ++i)
            stage[i] = *(const v2f*)(rep + cb + 4 * e_j[i]);
        #pragma unroll
        for (int i = 0; i < 4; ++i)
            lb[0][tx + 256 * i] = stage[i];
    }
    __syncthreads();

    for (int t = 0; t < TILES_PER_CHUNK; ++t) {
        const int ct  = tbase + t;
        const int cur = t & 1;

        // Issue next tile's global loads now; they retire under the WMMAs.
        if (t + 1 < TILES_PER_CHUNK) {
            const size_t cb = (size_t)((ct + 1) * 16 + e_col) * DDIM + e_kk;
            #pragma unroll
            for (int i = 0; i < 4; ++i)
                stage[i] = *(const v2f*)(rep + cb + 4 * e_j[i]);
        }

        // 16x16 tile of sim: 32 f32 WMMAs over K=128, two independent chains.
        v8f c0v = {}, c1v = {};
        #pragma unroll
        for (int j = 0; j < 32; j += 2) {
            v2f b0 = lb[cur][(j << 5) | lane];
            v2f b1 = lb[cur][((j + 1) << 5) | lane];
            c0v = __builtin_amdgcn_wmma_f32_16x16x4_f32(
                      false, a[j], false, b0, (short)0, c0v, false, false);
            c1v = __builtin_amdgcn_wmma_f32_16x16x4_f32(
                      false, a[j + 1], false, b1, (short)0, c1v, false, false);
        }

        // Commit staged registers to the alternate buffer, then one barrier.
        if (t + 1 < TILES_PER_CHUNK) {
            #pragma unroll
            for (int i = 0; i < 4; ++i)
                lb[cur ^ 1][tx + 256 * i] = stage[i];
        }
        __syncthreads();

        const bool isDiag = (ct == rt);
        const bool isPos  = (ct == posTile);

        // C layout: VGPR v -> row r0+v (lanes 0-15, col=lane) and
        //           row r0+v+8 (lanes 16-31, col=lane-16).
        #pragma unroll
        for (int v = 0; v < 8; ++v) {
            const float s = (c0v[v] + c1v[v]) * 2.0f;   // 1/TEMPERATURE
            if (isPos) {
                if (lane == v)      posArr[r0 + v]     = s;
                if (lane == v + 24) posArr[r0 + v + 8] = s;
            }
            float e = __expf(s);
            if (isDiag && (lane == v || lane == v + 24)) e = 0.0f;   // mask self
            acc[v] += e;
        }
    }

    // Reduce per-row exp-sums within 16-lane halves, combine across chunks.
    #pragma unroll
    for (int v = 0; v < 8; ++v) {
        float s = acc[v];
        #pragma unroll
        for (int off = 8; off >= 1; off >>= 1)
            s += __shfl_xor(s, off, 32);      // masks <16 keep halves separate
        if (lane == 0)  atomicAdd(&rowsum[r0 + v], s);
        if (lane == 16) atomicAdd(&rowsum[r0 + v + 8], s);
    }
}

// ---------------------------------------------------------------------------
// Kernel 3: loss = sum_i( log(rowsum[i]) - pos[i] ) / N
// ---------------------------------------------------------------------------
__global__ void ntx_finalize(const float* __restrict__ rowsum,
                             const float* __restrict__ posArr,
                             float* __restrict__ out) {
    __shared__ float red[256];
    float s = 0.0f;
    for (int i = threadIdx.x; i < NROWS; i += 256)
        s += __logf(rowsum[i]) - posArr[i];
    red[threadIdx.x] = s;
    __syncthreads();
    for (int st = 128; st >= 1; st >>= 1) {
        if ((int)threadIdx.x < st) red[threadIdx.x] += red[threadIdx.x + st];
        __syncthreads();
    }
    if (threadIdx.x == 0) out[0] = red[0] / (float)NROWS;
}

extern "C" void kernel_launch(void* const* d_in, const int* in_sizes, int n_in,
                              void* d_out, int out_size, void* d_ws, size_t ws_size,
                              hipStream_t stream) {
    const float* zis = (const float*)d_in[0];
    const float* zjs = (const float*)d_in[1];
    float* out = (float*)d_out;

    // Workspace layout: rep (4 MB) | rowsum (32 KB) | pos (32 KB)
    float* rep    = (float*)d_ws;
    float* rowsum = rep + (size_t)NROWS * DDIM;
    float* posArr = rowsum + NROWS;

    // 1) normalize: one wave per row -> 8192 waves = 1024 blocks x 256
    ntx_prep<<<dim3(1024), dim3(256), 0, stream>>>(zis, zjs, rep, rowsum);

    // 2) fused sim + streamed exp-sum: 64 row-blocks x 8 column-chunks
    ntx_main<<<dim3(64, 8), dim3(256), 0, stream>>>(rep, rowsum, posArr);

    // 3) scalar loss
    ntx_finalize<<<dim3(1), dim3(256), 0, stream>>>(rowsum, posArr, out);
}